// ROCKET_8667244003367
// MI455X (gfx1250) — compile-verified
//
#include <hip/hip_runtime.h>

#ifndef __has_builtin
#define __has_builtin(x) 0
#endif

#if __has_builtin(__builtin_amdgcn_global_load_async_to_lds_b32)
#define HAVE_ASYNC_LDS 1
#else
#define HAVE_ASYNC_LDS 0
#endif

#define AS1 __attribute__((address_space(1)))
#define AS3 __attribute__((address_space(3)))

static constexpr int BATCH    = 16;
static constexpr int C_IN     = 3;
static constexpr int SEQ      = 1000;
static constexpr int KMAX     = 11;
static constexpr int SLOTS    = SEQ + 2;   // 1002: zero-guard slot at 0 and 1001
static constexpr int NTHREADS = 512;       // 16 wave32: one wave per batch
static constexpr int KPB      = 4;         // kernels per block (amortize staging)

// LDS layout: smem4[b*SLOTS + (p+1)] = {x[b,0,p], x[b,1,p], x[b,2,p], pad}
// slot 0 and slot 1001 of each batch row are zeros (padding / OOB sentinel).
// Total: 16*1002*16 B = 250.5 KB  (needs CDNA5's 320 KB LDS per WGP).

__global__ __launch_bounds__(NTHREADS, 1)
void rocket_conv_feats(const float* __restrict__ x,
                       const float* __restrict__ weight,
                       const float* __restrict__ bias,
                       const int*   __restrict__ dil,
                       const int*   __restrict__ start,
                       const int*   __restrict__ olen,
                       const int*   __restrict__ pmaxp,
                       float*       __restrict__ out,
                       int nker)
{
    extern __shared__ float4 smem4[];
    float* smem_f = (float*)smem4;

    const int tid  = threadIdx.x;
    const int lane = tid & 31;
    const int b    = tid >> 5;              // wave id == batch id (16 waves)
    const int j0   = blockIdx.x * KPB;

    __builtin_prefetch(weight + (size_t)j0 * (C_IN * KMAX), 0, 1);  // global_prefetch_b8

    // Zero the guard slots (p = -1 and p = SEQ sentinel positions).
    if (tid < 2 * BATCH) {
        int gb   = tid >> 1;
        int slot = (tid & 1) ? (SLOTS - 1) : 0;
        smem4[gb * SLOTS + slot] = make_float4(0.f, 0.f, 0.f, 0.f);
    }

    // Stage x into LDS, transposing (b,c,p) -> [b][p+1][c] (channel-interleaved).
    // The async DMA's per-lane LDS destination address performs the transpose.
    for (int e = tid; e < BATCH * C_IN * SEQ; e += NTHREADS) {
        int bb = e / (C_IN * SEQ);
        int r  = e - bb * (C_IN * SEQ);
        int c  = r / SEQ;
        int p  = r - c * SEQ;
        int dst = (bb * SLOTS + p + 1) * 4 + c;   // float index
#if HAVE_ASYNC_LDS
        __builtin_amdgcn_global_load_async_to_lds_b32(
            (AS1 int*)(const_cast<float*>(x) + e),
            (AS3 int*)(smem_f + dst), 0, 0);
#else
        smem_f[dst] = x[e];
#endif
    }

#if HAVE_ASYNC_LDS
# if __has_builtin(__builtin_amdgcn_s_wait_asynccnt)
    __builtin_amdgcn_s_wait_asynccnt(0);
# else
    asm volatile("s_wait_asynccnt 0" ::: "memory");
# endif
#endif
    __syncthreads();

    const int pm = pmaxp[0];
    // Reference: idx = min(idx, L-1), L = SEQ + 2*pm, then x_pad lookup.
    // In slot space p = idx - pm: clamp to [-1, hi_slot]; hi_slot = SEQ when
    // pm>0 (slot SEQ+1 is a zero sentinel), SEQ-1 when pm==0 (must read x[999]).
    const int hi_slot = (pm == 0) ? (SEQ - 1) : SEQ;
    const int lo = b * SLOTS;               // slot of p = -1 (zero)
    const int hi = lo + hi_slot + 1;        // highest addressable slot
    const long long ncols = 2LL * nker;

#pragma unroll 1
    for (int jj = 0; jj < KPB; ++jj) {
        const int j = j0 + jj;
        if (j >= nker) break;

        // Per-kernel scalars (uniform across block -> SGPRs).
        const int   d  = dil[j];
        const int   s  = start[j];
        const int   ol = olen[j];
        const float bj = bias[j];
        const float* wj = weight + (size_t)j * (C_IN * KMAX);

        float wr[C_IN * KMAX];
#pragma unroll
        for (int i = 0; i < C_IN * KMAX; ++i) wr[i] = wj[i];  // uniform -> s_loads

        // Wave-uniform safe-interior bounds: a 32-wide t-tile [tb, tb+31] needs
        // no clamping iff  s+t-pm >= 0  and  s+t-pm+10*d <= SEQ-1  for all lanes.
        const int t_lo_safe = pm - s;
        const int t_hi_safe = (SEQ - 1) - (s - pm) - (KMAX - 1) * d;

        float maxv = -__builtin_inff();
        int   cnt  = 0;

        const int base0 = lo + (s - pm + 1);      // slot of (t=0, k=0)
        for (int tb = 0; tb < ol; tb += 32) {
            const int t  = tb + lane;
            float acc = bj;
            int   ik  = base0 + t;
            if (tb >= t_lo_safe && (tb + 31) <= t_hi_safe) {
                // Fast path: every tap of every lane hits real data.
#pragma unroll
                for (int k = 0; k < KMAX; ++k) {
                    float4 v = smem4[ik];             // ds_load_b128: 3 channels
                    acc = fmaf(v.x, wr[k],            acc);
                    acc = fmaf(v.y, wr[KMAX + k],     acc);
                    acc = fmaf(v.z, wr[2 * KMAX + k], acc);
                    ik += d;
                }
            } else {
                // Edge path: sentinel-clamped (exact reference semantics).
#pragma unroll
                for (int k = 0; k < KMAX; ++k) {
                    int a = (ik < lo) ? lo : ik;      // zero sentinel below
                    a     = (a > hi) ? hi : a;        // min(idx, L-1) / zero above
                    float4 v = smem4[a];
                    acc = fmaf(v.x, wr[k],            acc);
                    acc = fmaf(v.y, wr[KMAX + k],     acc);
                    acc = fmaf(v.z, wr[2 * KMAX + k], acc);
                    ik += d;
                }
            }
            if (t < ol) {
                maxv = fmaxf(maxv, acc);
                cnt += (acc > 0.0f) ? 1 : 0;
            }
        }

        // wave32 reductions (max and positive-count)
#pragma unroll
        for (int off = 16; off > 0; off >>= 1) {
            maxv = fmaxf(maxv, __shfl_xor(maxv, off, 32));
            cnt += __shfl_xor(cnt, off, 32);
        }

        if (lane == 0) {
            out[(long long)b * ncols + 2 * j]     = maxv;
            out[(long long)b * ncols + 2 * j + 1] = (float)cnt / (float)ol;
        }
    }
}

extern "C" void kernel_launch(void* const* d_in, const int* in_sizes, int n_in,
                              void* d_out, int out_size, void* d_ws, size_t ws_size,
                              hipStream_t stream)
{
    const float* x   = (const float*)d_in[0];
    const float* w   = (const float*)d_in[1];
    const float* b   = (const float*)d_in[2];
    const int*   dil = (const int*)d_in[3];
    const int*   st  = (const int*)d_in[4];
    const int*   ol  = (const int*)d_in[5];
    const int*   pm  = (const int*)d_in[6];
    float*       out = (float*)d_out;

    const int nker   = in_sizes[2];                      // 10000 kernels
    const int nblk   = (nker + KPB - 1) / KPB;           // 2500 blocks
    const size_t smem = (size_t)BATCH * SLOTS * sizeof(float4);  // 250.5 KB

    hipLaunchKernelGGL(rocket_conv_feats, dim3(nblk), dim3(NTHREADS), smem, stream,
                       x, w, b, dil, st, ol, pm, out, nker);
}